// RAAttention_47931835023493
// MI455X (gfx1250) — compile-verified
//
#include <hip/hip_runtime.h>
#include <hip/hip_bf16.h>

typedef float v2f __attribute__((ext_vector_type(2)));
typedef float v8f __attribute__((ext_vector_type(8)));

#define FDIM 64

// ---------- helpers ----------
__device__ __forceinline__ void atomicMaxFloat(float* addr, float val) {
    // order-preserving int punning; seg_max initialized to -inf
    if (val >= 0.0f) {
        atomicMax((int*)addr, __float_as_int(val));
    } else {
        atomicMin((unsigned int*)addr, __float_as_uint(val));
    }
}

// ---------- kernel 0: init output + softmax state ----------
__global__ void ra_init_kernel(const float* __restrict__ features,
                               float* __restrict__ out,
                               float* __restrict__ seg_max,
                               float* __restrict__ seg_sum,
                               int N) {
    int idx = blockIdx.x * blockDim.x + threadIdx.x;
    if (idx >= N * FDIM) return;
    int n = idx >> 6;
    int f = idx & 63;
    out[n * 2 * FDIM + f] = features[idx];          // concat left half
    out[n * 2 * FDIM + FDIM + f] = 0.0f;            // accumulator half
    if (f == 0) {
        seg_max[n] = -__builtin_inff();
        seg_sum[n] = 0.0f;
    }
}

// ---------- kernel 1: normalize rel_emb table + logit table via WMMA ----------
// One wave handles 16 relation rows. rt_table[r] = rel_emb[r]/||rel_emb[r]||,
// att_table[r] = rt_table[r] . w  via 16 chained V_WMMA_F32_16X16X4_F32
// (B[k][n] = w[k] broadcast over columns => every column of D is the logit vector).
__global__ void ra_relnorm_wmma_kernel(const float* __restrict__ rel_emb,
                                       const float* __restrict__ attn_w,
                                       float* __restrict__ rt_table,
                                       float* __restrict__ att_table,
                                       int R) {
    int lane = threadIdx.x & 31;
    int tile = blockIdx.x * (blockDim.x >> 5) + (threadIdx.x >> 5);
    if (tile * 16 >= R) return;               // wave-uniform exit: EXEC stays all-1s
    int base = tile * 16;
    int m  = lane & 15;                        // relation row within tile (A-row)
    int kh = lane >> 4;                        // K half
    int r  = base + m;
    const float* rrow = rel_emb + (size_t)r * FDIM;

    v2f a[16];
    float ss = 0.0f;
#pragma unroll
    for (int i = 0; i < 16; ++i) {
        float2 t = *(const float2*)(rrow + 4 * i + 2 * kh);
        v2f av; av.x = t.x; av.y = t.y;
        a[i] = av;
        ss += t.x * t.x + t.y * t.y;
    }
    ss += __shfl_xor(ss, 16, 32);             // pair the two K halves
    float scale = rsqrtf(fmaxf(ss, 1e-12f));

    v8f c = {};
#pragma unroll
    for (int i = 0; i < 16; ++i) {
        float2 wt = *(const float2*)(attn_w + 4 * i + 2 * kh);
        v2f b; b.x = wt.x; b.y = wt.y;
        v2f as; as.x = a[i].x * scale; as.y = a[i].y * scale;
        *(float2*)(rt_table + (size_t)r * FDIM + 4 * i + 2 * kh) =
            make_float2(as.x, as.y);
        c = __builtin_amdgcn_wmma_f32_16x16x4_f32(
                false, as, false, b, (short)0, c, false, false);
    }

    // D: lane l holds column N=l%16, rows M = v + 8*(l/16)
    if (m == 0) {
#pragma unroll
        for (int v = 0; v < 8; ++v) att_table[base + kh * 8 + v] = c[v];
    }
}

// ---------- kernel 2: per-edge logit lookup + segmented max (rows sorted) ----------
__global__ void ra_att_kernel(const float* __restrict__ att_table,
                              const float* __restrict__ sp_val,
                              const int*   __restrict__ adj,
                              const int*   __restrict__ spi,
                              float* __restrict__ att,
                              float* __restrict__ seg_max) {
    int e    = blockIdx.x * blockDim.x + threadIdx.x;
    int lane = threadIdx.x & 31;
    int ridx = spi[2 * e + 1];
    float sv = sp_val[e];
    // normalize(rel*sv) = sign(sv)*normalize(rel); sign only affects the logit
    float av = att_table[ridx] * (sv < 0.0f ? -1.0f : 1.0f);
    att[e] = av;
    int row = adj[2 * e];

    // segmented suffix-max over the wave
    float val = av;
#pragma unroll
    for (int off = 1; off < 32; off <<= 1) {
        float o    = __shfl_down(val, off, 32);
        int   orow = __shfl_down(row, off, 32);
        if (lane + off < 32 && orow == row) val = fmaxf(val, o);
    }
    int prow = __shfl_up(row, 1, 32);
    if (lane == 0 || prow != row) atomicMaxFloat(seg_max + row, val);
}

// ---------- kernel 3: exp + segmented denominator ----------
__global__ void ra_softmax_kernel(const float* __restrict__ att,
                                  const int*   __restrict__ adj,
                                  const float* __restrict__ seg_max,
                                  float* __restrict__ seg_sum,
                                  float* __restrict__ expv) {
    int e    = blockIdx.x * blockDim.x + threadIdx.x;
    int lane = threadIdx.x & 31;
    int row  = adj[2 * e];
    float ev = expf(att[e] - seg_max[row]);
    expv[e] = ev;

    float val = ev;
#pragma unroll
    for (int off = 1; off < 32; off <<= 1) {
        float o    = __shfl_down(val, off, 32);
        int   orow = __shfl_down(row, off, 32);
        if (lane + off < 32 && orow == row) val += o;
    }
    int prow = __shfl_up(row, 1, 32);
    if (lane == 0 || prow != row) atomicAdd(seg_sum + row, val);
}

// ---------- kernel 3b: reciprocal of denominators ----------
__global__ void ra_recip_kernel(float* __restrict__ seg_sum, int N) {
    int i = blockIdx.x * blockDim.x + threadIdx.x;
    if (i < N) seg_sum[i] = 1.0f / seg_sum[i];   // inf for edgeless rows: never read
}

// ---------- kernel 4: reflect, weight, run-length scatter ----------
// 16 lanes per edge (float4/lane): two edges in flight per wave, b128 loads,
// 4-step half-wave reduction for the projection dot.
__global__ void ra_scatter_kernel(const float* __restrict__ features,
                                  const float* __restrict__ rt_table,
                                  const int*   __restrict__ adj,
                                  const int*   __restrict__ spi,
                                  const float* __restrict__ expv,
                                  const float* __restrict__ inv_sum,
                                  float* __restrict__ out) {
    const int CHUNK = 32;                    // edges per wave, 16 per half-wave
    int lane = threadIdx.x & 31;
    int half = lane >> 4;
    int hl   = lane & 15;
    int wid  = blockIdx.x * (blockDim.x >> 5) + (threadIdx.x >> 5);
    int base = wid * CHUNK + half * (CHUNK / 2);

    float ax = 0.f, ay = 0.f, az = 0.f, aw = 0.f;
    int cur = -1;

    for (int i = 0; i < CHUNK / 2; ++i) {
        int e    = base + i;
        int row  = adj[2 * e];
        int col  = adj[2 * e + 1];
        int ridx = spi[2 * e + 1];
        float alpha = expv[e] * inv_sum[row];

        float4 rt = *(const float4*)(rt_table + (size_t)ridx * FDIM + 4 * hl);
        float4 nb = *(const float4*)(features + (size_t)col  * FDIM + 4 * hl);

        float pp = nb.x * rt.x + nb.y * rt.y + nb.z * rt.z + nb.w * rt.w;
#pragma unroll
        for (int off = 8; off >= 1; off >>= 1)  // xor<16: stays within half-wave
            pp += __shfl_xor(pp, off, 32);
        float s = 2.0f * pp;

        float cx = (nb.x - s * rt.x) * alpha;
        float cy = (nb.y - s * rt.y) * alpha;
        float cz = (nb.z - s * rt.z) * alpha;
        float cw = (nb.w - s * rt.w) * alpha;

        if (row != cur) {
            if (cur >= 0) {
                float* p = out + (size_t)cur * 2 * FDIM + FDIM + 4 * hl;
                atomicAdd(p + 0, ax); atomicAdd(p + 1, ay);
                atomicAdd(p + 2, az); atomicAdd(p + 3, aw);
            }
            cur = row; ax = cx; ay = cy; az = cz; aw = cw;
        } else {
            ax += cx; ay += cy; az += cz; aw += cw;
        }
    }
    if (cur >= 0) {
        float* p = out + (size_t)cur * 2 * FDIM + FDIM + 4 * hl;
        atomicAdd(p + 0, ax); atomicAdd(p + 1, ay);
        atomicAdd(p + 2, az); atomicAdd(p + 3, aw);
    }
}

// ---------- launcher ----------
extern "C" void kernel_launch(void* const* d_in, const int* in_sizes, int n_in,
                              void* d_out, int out_size, void* d_ws, size_t ws_size,
                              hipStream_t stream) {
    const float* features = (const float*)d_in[0];
    const float* rel_emb  = (const float*)d_in[1];
    const float* attn_w   = (const float*)d_in[2];
    const float* sp_val   = (const float*)d_in[3];
    const int*   adj      = (const int*)d_in[4];
    const int*   spi      = (const int*)d_in[5];
    float*       out      = (float*)d_out;

    int E = in_sizes[3];            // 1,600,000
    int N = in_sizes[0] / FDIM;     // 100,000
    int R = in_sizes[1] / FDIM;     // 2,000

    // workspace carve-out
    float* att       = (float*)d_ws;        // E
    float* expv      = att + E;             // E
    float* seg_max   = expv + E;            // N
    float* seg_sum   = seg_max + N;         // N
    float* rt_table  = seg_sum + N;         // R*64
    float* att_table = rt_table + (size_t)R * FDIM;  // R

    ra_init_kernel<<<(N * FDIM + 255) / 256, 256, 0, stream>>>(
        features, out, seg_max, seg_sum, N);

    int tiles = (R + 15) / 16;              // 125 for R=2000
    ra_relnorm_wmma_kernel<<<(tiles + 7) / 8, 256, 0, stream>>>(
        rel_emb, attn_w, rt_table, att_table, R);

    ra_att_kernel<<<E / 256, 256, 0, stream>>>(
        att_table, sp_val, adj, spi, att, seg_max);

    ra_softmax_kernel<<<E / 256, 256, 0, stream>>>(
        att, adj, seg_max, seg_sum, expv);

    ra_recip_kernel<<<(N + 255) / 256, 256, 0, stream>>>(seg_sum, N);

    int waves = E / 32;                     // 50,000
    ra_scatter_kernel<<<waves / 8, 256, 0, stream>>>(
        features, rt_table, adj, spi, expv, seg_sum, out);
}